// LISTA_22144851378559
// MI455X (gfx1250) — compile-verified
//
#include <hip/hip_runtime.h>
#include <cstdint>

// ---------------------------------------------------------------------------
// LISTA unrolled (3 stages) for MI455X / gfx1250.
//   conv5x5 (1ch, SAME) as 25 banded V_WMMA_F32_16X16X4_F32 per 16x16 tile.
//   BatchNorm batch stats via per-WG partials + 1-block finalize (no atomics).
//   Stage pre-op (affine BN + shrinkage [+ c0]) fused into the halo staging.
// Workspace layout (needs ~134.3 MB):
//   [0, 64MB)        t-scratch (t0, later t2)
//   [64MB, 128MB)    c0 = BN(conv(y))
//   [128MB, +64KB)   per-workgroup float2 partials (sum, sumsq)
//   then             6 floats of affine stats (a,b per stage)
// d_out is reused as the t1 buffer (fully written before being read).
// ---------------------------------------------------------------------------

typedef __attribute__((ext_vector_type(8))) float v8f;
typedef __attribute__((ext_vector_type(2))) float v2f;

#define WIMG 512
#define HIMG 512
#define NIMG 64
#define TILE_R 16
#define WG_COLS 128
#define PATCH_R 20
#define PATCH_C 132
#define NG_X (WIMG / WG_COLS)        /* 4  col-groups per row of tiles */
#define NG_R (HIMG / TILE_R)         /* 32 row-tiles per image         */
#define NGROUPS (NIMG * NG_R * NG_X) /* 8192 workgroups                */
#define TOTAL_ELEMS (64ll * 512 * 512)
#define BN_EPS 1e-5f

#if defined(__AMDGCN__) && __has_builtin(__builtin_amdgcn_wmma_f32_16x16x4_f32)
#define USE_WMMA_F32 1
#else
#define USE_WMMA_F32 0
#if defined(__AMDGCN__)
#warning "gfx1250: __builtin_amdgcn_wmma_f32_16x16x4_f32 not found; VALU fallback in use"
#endif
#endif

#if defined(__AMDGCN__) &&                                                     \
    __has_builtin(__builtin_amdgcn_global_load_async_to_lds_b32) &&            \
    __has_builtin(__builtin_amdgcn_s_wait_asynccnt)
#define USE_ASYNC_LDS 1
typedef __attribute__((address_space(1))) int as1_int; // global (addrspace 1)
typedef __attribute__((address_space(3))) int as3_int; // LDS    (addrspace 3)
#else
#define USE_ASYNC_LDS 0
#if defined(__AMDGCN__)
#warning "gfx1250: async global->LDS builtins not found; plain staging in use"
#endif
#endif

__device__ __forceinline__ float shrinkf(float x, float lam) {
  const float d1 = x - lam, d2 = x + lam;
  return x + 0.5f * (__builtin_sqrtf(__builtin_fmaf(d1, d1, 1.f)) -
                     __builtin_sqrtf(__builtin_fmaf(d2, d2, 1.f)));
}

// 5-way select of uniform scalars by per-lane index; returns 0 out of range.
__device__ __forceinline__ float sel5(float w0, float w1, float w2, float w3,
                                      float w4, int i) {
  float r = 0.f;
  r = (i == 0) ? w0 : r;
  r = (i == 1) ? w1 : r;
  r = (i == 2) ? w2 : r;
  r = (i == 3) ? w3 : r;
  r = (i == 4) ? w4 : r;
  return r;
}

// MODE 0: x = y                       (conv input is raw y)
// MODE 1: x = shrink(a*t0+b, lam);    also writes c0 = a*t0+b (interior)
// MODE 2: x = shrink(a*t1+b + c0, lam)
template <int MODE>
__global__ __launch_bounds__(256) void conv_bn_stage(
    const float* __restrict__ xsrc, const float* __restrict__ tprev,
    const float* __restrict__ c0in, const float* __restrict__ stats,
    const float* __restrict__ lambd, int lidx, const float* __restrict__ convw,
    const float* __restrict__ convb, float* __restrict__ tout,
    float* __restrict__ c0out, float2* __restrict__ partials) {
  __shared__ __align__(16) float patch[PATCH_R * PATCH_C];
  __shared__ float red[512];

  const int g = blockIdx.x;
  const int cg = g % NG_X;
  const int tr = (g / NG_X) % NG_R;
  const int img = g / (NG_X * NG_R);
  const int tid = threadIdx.x;
  const long imgoff = (long)img * (HIMG * WIMG);
  const int pr0 = tr * TILE_R - 2;  // patch origin (global row)
  const int pc0 = cg * WG_COLS - 2; // patch origin (global col)

  float aff_a = 1.f, aff_b = 0.f, lam = 0.f;
  if (MODE >= 1) {
    aff_a = stats[0];
    aff_b = stats[1];
    lam = lambd[lidx];
  }

  // ---- stage 20x132 halo patch into LDS, pre-op applied pointwise ----
  for (int idx = tid; idx < PATCH_R * PATCH_C; idx += 256) {
    const int pr = idx / PATCH_C;
    const int pc = idx - pr * PATCH_C;
    const int r = pr0 + pr, c = pc0 + pc;
    const bool inb = (r >= 0) & (r < HIMG) & (c >= 0) & (c < WIMG);
    if (MODE == 0) {
#if USE_ASYNC_LDS
      if (inb) {
        const float* gsrc = xsrc + imgoff + (long)r * WIMG + c;
        __builtin_amdgcn_global_load_async_to_lds_b32(
            (as1_int*)(uintptr_t)gsrc,
            (as3_int*)(uint32_t)(uintptr_t)&patch[idx], 0, 0);
      } else {
        patch[idx] = 0.f;
      }
#else
      patch[idx] = inb ? xsrc[imgoff + (long)r * WIMG + c] : 0.f;
#endif
    } else {
      float x = 0.f;
      if (inb) {
        const long p = imgoff + (long)r * WIMG + c;
        float bn = __builtin_fmaf(aff_a, tprev[p], aff_b);
        if (MODE == 2) bn += c0in[p];
        if (MODE == 1 && pr >= 2 && pr < 2 + TILE_R && pc >= 2 &&
            pc < 2 + WG_COLS)
          c0out[p] = bn; // each pixel is interior of exactly one WG
        x = shrinkf(bn, lam);
      }
      patch[idx] = x;
    }
  }
#if USE_ASYNC_LDS
  if (MODE == 0) __builtin_amdgcn_s_wait_asynccnt(0);
#endif
  __syncthreads();

  // ---- uniform conv weights ----
  float w[25];
#pragma unroll
  for (int i = 0; i < 25; ++i) w[i] = convw[i];
  const float bias = convb[0];

  const int lane = tid & 31;
  const int wave = tid >> 5; // 8 waves -> 8 adjacent 16x16 tiles
  const int m = lane & 15;   // A: M row / B,C,D: N column-of-lane (ISA 7.12.2)
  const int kb = (lane >> 4) << 1; // K pair base: lanes 0-15 -> {0,1}, 16-31 -> {2,3}
  const int n = m;                 // output row within tile for this lane
  const int waveoff = wave * 16;   // tile column offset within WG region

  v8f acc = {bias, bias, bias, bias, bias, bias, bias, bias};

#if USE_WMMA_F32
  // Banded A fragments: A_{dr,j}[m][k] = w[dr][4j+k-m] (0 outside the band).
  float A0[25], A1[25];
#pragma unroll
  for (int dri = 0; dri < 5; ++dri) {
#pragma unroll
    for (int j = 0; j < 5; ++j) {
      const int i0 = 4 * j + kb - m;
      const float* wr = &w[dri * 5];
      A0[dri * 5 + j] = sel5(wr[0], wr[1], wr[2], wr[3], wr[4], i0);
      A1[dri * 5 + j] = sel5(wr[0], wr[1], wr[2], wr[3], wr[4], i0 + 1);
    }
  }
  // 25 chained V_WMMA_F32_16X16X4_F32; B frag = one ds_load_b64 per lane.
#pragma unroll
  for (int dri = 0; dri < 5; ++dri) {
#pragma unroll
    for (int j = 0; j < 5; ++j) {
      const int off = (n + dri) * PATCH_C + waveoff + 4 * j + kb;
      v2f B = *(const v2f*)&patch[off];
      v2f A = {A0[dri * 5 + j], A1[dri * 5 + j]};
      acc = __builtin_amdgcn_wmma_f32_16x16x4_f32(false, A, false, B, (short)0,
                                                  acc, false, false);
    }
  }
#else
  // VALU fallback: same lane->pixel mapping as the WMMA C/D layout.
#pragma unroll
  for (int dri = 0; dri < 5; ++dri) {
#pragma unroll
    for (int dc = 0; dc < 5; ++dc) {
      const float wv = w[dri * 5 + dc];
#pragma unroll
      for (int i = 0; i < 8; ++i) {
        const int cl = waveoff + ((lane >> 4) << 3) + i;
        acc[i] = __builtin_fmaf(wv, patch[(n + dri) * PATCH_C + cl + dc], acc[i]);
      }
    }
  }
#endif

  // ---- store conv tile: lane covers row n, 8 contiguous cols (2x b128) ----
  {
    const int outr = tr * TILE_R + n;
    const int outc = cg * WG_COLS + waveoff + ((lane >> 4) << 3);
    float* dst = tout + imgoff + (long)outr * WIMG + outc;
    *(float4*)dst = make_float4(acc[0], acc[1], acc[2], acc[3]);
    *(float4*)(dst + 4) = make_float4(acc[4], acc[5], acc[6], acc[7]);
  }

  // ---- per-WG batch-stat partials (fixed-order tree, deterministic) ----
  float s = 0.f, s2 = 0.f;
#pragma unroll
  for (int i = 0; i < 8; ++i) {
    s += acc[i];
    s2 = __builtin_fmaf(acc[i], acc[i], s2);
  }
  red[tid] = s;
  red[256 + tid] = s2;
  __syncthreads();
#pragma unroll
  for (int off = 128; off > 0; off >>= 1) {
    if (tid < off) {
      red[tid] += red[tid + off];
      red[256 + tid] += red[256 + tid + off];
    }
    __syncthreads();
  }
  if (tid == 0) partials[g] = make_float2(red[0], red[256]);
}

__global__ __launch_bounds__(256) void finalize_stats(
    const float2* __restrict__ partials, const float* __restrict__ gamma,
    const float* __restrict__ beta, float* __restrict__ stats) {
  __shared__ float red[512];
  const int tid = threadIdx.x;
  float s = 0.f, s2 = 0.f;
  for (int i = tid; i < NGROUPS; i += 256) {
    s += partials[i].x;
    s2 += partials[i].y;
  }
  red[tid] = s;
  red[256 + tid] = s2;
  __syncthreads();
  for (int off = 128; off > 0; off >>= 1) {
    if (tid < off) {
      red[tid] += red[tid + off];
      red[256 + tid] += red[256 + tid + off];
    }
    __syncthreads();
  }
  if (tid == 0) {
    const float inv = 1.f / (float)TOTAL_ELEMS;
    const float mean = red[0] * inv;
    const float var = red[256] * inv - mean * mean;
    const float a = gamma[0] * __frsqrt_rn(var + BN_EPS);
    stats[0] = a;
    stats[1] = __builtin_fmaf(-mean, a, beta[0]);
  }
}

__global__ __launch_bounds__(256) void final_stage(
    const float4* __restrict__ t2, const float4* __restrict__ c0,
    const float* __restrict__ stats, const float* __restrict__ lambd,
    float4* __restrict__ out) {
  const float a = stats[0], b = stats[1], lam = lambd[2];
  const long i = (long)blockIdx.x * 256 + threadIdx.x;
  const float4 t = t2[i];
  const float4 c = c0[i];
  float4 r;
  r.x = shrinkf(__builtin_fmaf(a, t.x, b) + c.x, lam);
  r.y = shrinkf(__builtin_fmaf(a, t.y, b) + c.y, lam);
  r.z = shrinkf(__builtin_fmaf(a, t.z, b) + c.z, lam);
  r.w = shrinkf(__builtin_fmaf(a, t.w, b) + c.w, lam);
  out[i] = r;
}

extern "C" void kernel_launch(void* const* d_in, const int* in_sizes, int n_in,
                              void* d_out, int out_size, void* d_ws,
                              size_t ws_size, hipStream_t stream) {
  (void)in_sizes;
  (void)n_in;
  (void)out_size;
  (void)ws_size;
  const float* y = (const float*)d_in[0];
  const float* convw = (const float*)d_in[1];
  const float* convb = (const float*)d_in[2];
  const float* gamma = (const float*)d_in[3];
  const float* beta = (const float*)d_in[4];
  const float* lambd = (const float*)d_in[5];
  float* out = (float*)d_out;

  char* ws = (char*)d_ws;
  float* tbuf = (float*)ws;                                       // t0 / t2
  float* c0b = (float*)(ws + TOTAL_ELEMS * sizeof(float));        // c0
  float2* partials = (float2*)(ws + 2 * TOTAL_ELEMS * sizeof(float));
  float* stats = (float*)(ws + 2 * TOTAL_ELEMS * sizeof(float) +
                          NGROUPS * sizeof(float2));

  const dim3 blk(256);
  const dim3 grd(NGROUPS);

  // Stage 1: t0 = conv(y); stats0
  conv_bn_stage<0><<<grd, blk, 0, stream>>>(y, nullptr, nullptr, nullptr,
                                            lambd, 0, convw, convb, tbuf,
                                            nullptr, partials);
  finalize_stats<<<1, blk, 0, stream>>>(partials, gamma, beta, stats + 0);

  // Stage 2: c0 = BN(t0); x1 = shrink(c0, l0); t1 = conv(x1) -> d_out; stats1
  conv_bn_stage<1><<<grd, blk, 0, stream>>>(nullptr, tbuf, nullptr, stats + 0,
                                            lambd, 0, convw, convb, out, c0b,
                                            partials);
  finalize_stats<<<1, blk, 0, stream>>>(partials, gamma, beta, stats + 2);

  // Stage 3: x2 = shrink(BN(t1)+c0, l1); t2 = conv(x2) -> tbuf; stats2
  conv_bn_stage<2><<<grd, blk, 0, stream>>>(nullptr, out, c0b, stats + 2,
                                            lambd, 1, convw, convb, tbuf,
                                            nullptr, partials);
  finalize_stats<<<1, blk, 0, stream>>>(partials, gamma, beta, stats + 4);

  // Final: out = shrink(BN(t2)+c0, l2)
  final_stage<<<dim3((unsigned)(TOTAL_ELEMS / 4 / 256)), blk, 0, stream>>>(
      (const float4*)tbuf, (const float4*)c0b, stats + 4, lambd, (float4*)out);
}